// DynamicTopGate_27453430956611
// MI455X (gfx1250) — compile-verified
//
#include <hip/hip_runtime.h>
#include <cstdint>

// Problem constants from the reference
#define B_ROWS   16384
#define IN_DIM   2048
#define E_DIM    16
#define KMAX_OUT 8

typedef float v2f __attribute__((ext_vector_type(2)));
typedef float v4f __attribute__((ext_vector_type(4)));
typedef float v8f __attribute__((ext_vector_type(8)));

// Block = 8 waves = 4 row-tiles x 2 K-halves (K-split doubles occupancy for MLP).
//   GEMM1: h = tanh(x @ W1^T) via V_WMMA_F32_16X16X4_F32, each wave K=1024,
//          partner halves reduced through LDS.
//   GEMM2: logits = h @ W2^T / 0.7 via 4 WMMAs (A restaged through LDS).
//   Gate:  warp-parallel rank-based stable argsort + softmax + k rules.
__global__ __launch_bounds__(256)
void DynamicTopGate_27453430956611_kernel(const float* __restrict__ x,
                                          const float* __restrict__ W1,
                                          const float* __restrict__ W2,
                                          float* __restrict__ out)
{
    __shared__ __align__(16) float ldsAcc[4][32][8];   // partner K-half accumulators
    __shared__ __align__(16) float ldsH[4][256];       // per-tile 16x16 h tile
    __shared__ __align__(16) float ldsL[4][256];       // per-tile 16x16 logits tile
    __shared__ __align__(16) float ldsP[4][16][8];     // per-row top-8 probs (sorted)
    __shared__ __align__(16) float ldsI[4][16][8];     // per-row top-8 indices (sorted)

    const int lane  = threadIdx.x & 31;
    const int wave  = threadIdx.x >> 5;
    const int tile  = wave & 3;             // 4 tiles per block
    const int kStart = (wave >> 2) * (IN_DIM / 2);  // 0 or 1024
    const int laneM = lane & 15;            // row (A) / col (B) index
    const int kHalf = (lane >> 4) << 1;     // +0 for lanes 0-15, +2 for 16-31
    const int tileRow = (blockIdx.x * 4 + tile) * 16;

    // A: lane holds (x[row][k+kHalf], x[row][k+kHalf+1])
    const float* xp  = x  + (size_t)(tileRow + laneM) * IN_DIM + kStart + kHalf;
    // B[k][n] = W1[n][k]; lane holds (W1[n][k+kHalf], W1[n][k+kHalf+1])
    const float* w1p = W1 + (size_t)laneM * IN_DIM + kStart + kHalf;

    // ---- GEMM1: this wave's K-half, K-step 4, unrolled x16 ----
    v8f acc = {};
    for (int kk = 0; kk < IN_DIM / 2; kk += 64) {
        // cover the full 256B/row consumed per iteration, 4KB ahead (speculative)
        __builtin_prefetch(xp + kk + 1024,      0, 3);   // global_prefetch_b8
        __builtin_prefetch(xp + kk + 1024 + 32, 0, 3);   // +128B line
#pragma unroll
        for (int k0 = 0; k0 < 64; k0 += 4) {
            v2f a = *(const v2f*)(xp  + kk + k0);
            v2f b = *(const v2f*)(w1p + kk + k0);
            acc = __builtin_amdgcn_wmma_f32_16x16x4_f32(
                false, a, false, b, (short)0, acc, false, false);
        }
    }

    // ---- K-split reduction: waves 4-7 publish, waves 0-3 accumulate ----
    if (wave >= 4) {
        v4f a0 = { acc[0], acc[1], acc[2], acc[3] };
        v4f a1 = { acc[4], acc[5], acc[6], acc[7] };
        *(v4f*)(&ldsAcc[tile][lane][0]) = a0;
        *(v4f*)(&ldsAcc[tile][lane][4]) = a1;
    }
    __syncthreads();

    const int mBase = (lane >> 4) << 3;     // D layout: VGPR j -> M = j (+8 hi half)

    if (wave < 4) {
        v4f p0 = *(const v4f*)(&ldsAcc[tile][lane][0]);
        v4f p1 = *(const v4f*)(&ldsAcc[tile][lane][4]);
#pragma unroll
        for (int j = 0; j < 4; ++j) { acc[j] += p0[j]; acc[j + 4] += p1[j]; }

        // ---- tanh, stage h tile to LDS in row-major [m][e] ----
#pragma unroll
        for (int j = 0; j < 8; ++j)
            ldsH[tile][(j + mBase) * 16 + laneM] = tanhf(acc[j]);
    }
    __syncthreads();

    if (wave < 4) {
        // ---- GEMM2: logits = h @ W2^T, K=16 (4 WMMAs), A from LDS ----
        v8f acc2 = {};
#pragma unroll
        for (int k0 = 0; k0 < 16; k0 += 4) {
            v2f a = *(const v2f*)(&ldsH[tile][laneM * 16 + k0 + kHalf]);
            v2f b = *(const v2f*)(W2 + laneM * 16 + k0 + kHalf);
            acc2 = __builtin_amdgcn_wmma_f32_16x16x4_f32(
                false, a, false, b, (short)0, acc2, false, false);
        }
#pragma unroll
        for (int j = 0; j < 8; ++j)
            ldsL[tile][(j + mBase) * 16 + laneM] = acc2[j] / 0.7f;   // / TEMP
    }
    __syncthreads();

    // ---- Phase A: warp-parallel rank + softmax. 2 lanes/row, 8 elems/lane.
    // rank(e) = #{j: v_j > v_e} + #{j < e: v_j == v_e}  ==  stable desc argsort.
    if (wave < 4) {
        const int rRow  = lane >> 1;           // 0..15
        const int eBase = (lane & 1) << 3;     // 0 or 8

        float rv[16];
#pragma unroll
        for (int q = 0; q < 4; ++q) {
            v4f t = *(const v4f*)(&ldsL[tile][rRow * 16 + q * 4]);
            rv[q * 4 + 0] = t[0]; rv[q * 4 + 1] = t[1];
            rv[q * 4 + 2] = t[2]; rv[q * 4 + 3] = t[3];
        }
        float mx = rv[0];
#pragma unroll
        for (int e = 1; e < 16; ++e) mx = fmaxf(mx, rv[e]);
        float ex[16]; float s = 0.0f;
#pragma unroll
        for (int e = 0; e < 16; ++e) { ex[e] = expf(rv[e] - mx); s += ex[e]; }
        const float inv = 1.0f / s;

#pragma unroll
        for (int t = 0; t < 8; ++t) {
            const int e = eBase + t;
            const float ve = rv[e];
            int rank = 0;
#pragma unroll
            for (int j = 0; j < 16; ++j)
                rank += ((rv[j] > ve) || ((rv[j] == ve) && (j < e))) ? 1 : 0;
            if (rank < KMAX_OUT) {                       // only top-8 matter
                ldsP[tile][rRow][rank] = ex[e] * inv;
                ldsI[tile][rRow][rank] = (float)e;
            }
        }
    }
    __syncthreads();

    // ---- Phase B: lanes 0..15 of waves 0-3 finalize one row each ----
    if (wave < 4 && lane < 16) {
        const int row = tileRow + lane;
        v4f pa = *(const v4f*)(&ldsP[tile][lane][0]);
        v4f pb = *(const v4f*)(&ldsP[tile][lane][4]);
        v4f ia = *(const v4f*)(&ldsI[tile][lane][0]);
        v4f ib = *(const v4f*)(&ldsI[tile][lane][4]);

        const float c0 = pa[0];
        const float c1 = c0 + pa[1];
        const float c2 = c1 + pa[2];
        // first position where cumsum >= P_MIN (anything past 3 clips to 3 anyway)
        int kv = (c0 >= 0.92f) ? 1 : (c1 >= 0.92f) ? 2 : (c2 >= 0.92f) ? 3 : 16;
        if (pa[0] >= 0.46f && (pa[0] - pa[1]) >= 0.1f) kv = 1;            // confident_1
        if (kv > 2 && (c1 >= (0.92f - 0.1f) || pa[2] <= 0.12f ||
                       (pa[1] - pa[2]) <= 0.03f)) kv = 2;                 // demote
        kv = kv > 3 ? 3 : kv;                                             // clip [1,3]

        v4f sa, sb, ma, mb;
#pragma unroll
        for (int j = 0; j < 4; ++j) {
            bool selA = j < kv;
            bool selB = (j + 4) < kv;
            sa[j] = selA ? pa[j] : 0.0f;  ma[j] = selA ? 1.0f : 0.0f;
            sb[j] = selB ? pb[j] : 0.0f;  mb[j] = selB ? 1.0f : 0.0f;
        }

        float* oIdx   = out;
        float* oScore = out + (size_t)B_ROWS * KMAX_OUT;
        float* oMask  = out + (size_t)2 * B_ROWS * KMAX_OUT;
        float* oK     = out + (size_t)3 * B_ROWS * KMAX_OUT;
        const size_t ro = (size_t)row * KMAX_OUT;

        *(v4f*)(oIdx   + ro)     = ia;  *(v4f*)(oIdx   + ro + 4) = ib;
        *(v4f*)(oScore + ro)     = sa;  *(v4f*)(oScore + ro + 4) = sb;
        *(v4f*)(oMask  + ro)     = ma;  *(v4f*)(oMask  + ro + 4) = mb;
        oK[row] = (float)kv;
    }
}

extern "C" void kernel_launch(void* const* d_in, const int* in_sizes, int n_in,
                              void* d_out, int out_size, void* d_ws, size_t ws_size,
                              hipStream_t stream)
{
    (void)in_sizes; (void)n_in; (void)out_size; (void)d_ws; (void)ws_size;
    const float* x  = (const float*)d_in[0];
    const float* W1 = (const float*)d_in[1];
    const float* W2 = (const float*)d_in[2];
    float* out = (float*)d_out;

    // 8 waves/block = 4 tiles x 2 K-halves -> 64 rows/block -> 256 blocks
    dim3 grid(B_ROWS / 64);
    dim3 block(256);
    DynamicTopGate_27453430956611_kernel<<<grid, block, 0, stream>>>(x, W1, W2, out);
}